// E2EMN_50792283242815
// MI455X (gfx1250) — compile-verified
//
#include <hip/hip_runtime.h>
#include <math.h>

#define VOCAB 50257
#define EMBED 128
#define NSENT 50
#define TOK   20
#define BATCH 64
#define NHOPS 3

typedef __attribute__((ext_vector_type(2))) float v2f;
typedef __attribute__((ext_vector_type(8))) float v8f;

// ---------------------------------------------------------------------------
// u[b,d] = sum_t W_B[questions[b,t], d]
__global__ void k_embed_q(const int* __restrict__ questions,
                          const float* __restrict__ W_B,
                          float* __restrict__ u) {
  const int b = blockIdx.x;
  const int d = threadIdx.x;            // 0..127
  float s = 0.f;
#pragma unroll
  for (int t = 0; t < TOK; ++t) {
    const int tok = questions[b * TOK + t];
    s += W_B[tok * EMBED + d];
  }
  u[b * EMBED + d] = s;
}

// ---------------------------------------------------------------------------
// m[b,s,d] = sum_t W_A[story[b,s,t], d] + W_AT[te(b,s), d]   (same for c / W_C / W_CT)
// te(b,s) = s+1 if any mask token == 0 else 0
__global__ void k_embed_mc(const int* __restrict__ stories,
                           const int* __restrict__ masks,
                           const float* __restrict__ W_A,
                           const float* __restrict__ W_C,
                           const float* __restrict__ W_AT,
                           const float* __restrict__ W_CT,
                           float* __restrict__ m,
                           float* __restrict__ c) {
  const int blk = blockIdx.x;           // b*NSENT + s
  const int s   = blk % NSENT;
  const int d   = threadIdx.x;          // 0..127
  const int* story = stories + blk * TOK;
  const int* msk   = masks   + blk * TOK;
  int wl = 0;
#pragma unroll
  for (int t = 0; t < TOK; ++t) wl += (msk[t] == 0) ? 1 : 0;
  const int te = (wl >= 1) ? (s + 1) : 0;
  float ms = W_AT[te * EMBED + d];
  float cs = W_CT[te * EMBED + d];
#pragma unroll
  for (int t = 0; t < TOK; ++t) {
    const int tok = story[t];
    ms += W_A[tok * EMBED + d];
    cs += W_C[tok * EMBED + d];
  }
  m[blk * EMBED + d] = ms;
  c[blk * EMBED + d] = cs;
}

// ---------------------------------------------------------------------------
// 3 hops: score = m.u ; softmax ; u += c^T softmax(score)
__global__ void k_hops(const float* __restrict__ m,
                       const float* __restrict__ c,
                       float* __restrict__ u) {
  __shared__ float u_s[EMBED];
  __shared__ float score[NSENT];
  const int b = blockIdx.x;
  const int t = threadIdx.x;            // 0..127
  const float* mb = m + b * NSENT * EMBED;
  const float* cb = c + b * NSENT * EMBED;
  u_s[t] = u[b * EMBED + t];
  __syncthreads();

  for (int hop = 0; hop < NHOPS; ++hop) {
    if (t < NSENT) {
      float acc = 0.f;
      const float* row = mb + t * EMBED;
#pragma unroll 8
      for (int d = 0; d < EMBED; ++d) acc += row[d] * u_s[d];
      score[t] = acc;
    }
    __syncthreads();
    // redundant per-thread max (50 elems, trivial)
    float mx = -1e30f;
    for (int s2 = 0; s2 < NSENT; ++s2) mx = fmaxf(mx, score[s2]);
    __syncthreads();
    if (t < NSENT) score[t] = __expf(score[t] - mx);
    __syncthreads();
    float ssum = 0.f;
    for (int s2 = 0; s2 < NSENT; ++s2) ssum += score[s2];
    const float inv = 1.f / ssum;
    float o = 0.f;
    for (int s2 = 0; s2 < NSENT; ++s2) o += score[s2] * cb[s2 * EMBED + t];
    __syncthreads();                    // done reading score before next hop rewrites it
    u_s[t] += o * inv;
    __syncthreads();
  }
  u[b * EMBED + t] = u_s[t];
}

// ---------------------------------------------------------------------------
// out[64, VOCAB] = u[64,128] @ W_lin^T + b_lin   via V_WMMA_F32_16X16X4_F32.
// One wave per 16-column N-tile, computing ALL 4 M-tiles (M=64) so each
// W_lin B-fragment is loaded once and reused by 4 WMMAs (4x less L2 panel
// traffic on the dominant 25.7 MB operand). 32 k-steps of K=4 -> 128 WMMAs.
// A layout (16x4 f32): lane L<16 -> M=L, VGPR0=K0, VGPR1=K1; L>=16 -> M=L-16, K2/K3.
// B layout (4x16 f32): mirrored, lane%16 = N column (W_lin row), same K split.
// C/D layout: VGPR r, lane L -> M = r + 8*(L>=16), N = L%16.
__global__ void k_gemm(const float* __restrict__ u,
                       const float* __restrict__ W_lin,
                       const float* __restrict__ b_lin,
                       float* __restrict__ out) {
  const int NT   = (VOCAB + 15) / 16;                 // 3142 N tiles
  const int wave = (int)((blockIdx.x * blockDim.x + threadIdx.x) >> 5);
  const int lane = (int)(threadIdx.x & 31);
  if (wave >= NT) return;                             // wave-uniform branch
  const int nbase = wave * 16;
  const int half  = lane >> 4;                        // K sub-pair select
  const int l16   = lane & 15;

  const int nrow   = nbase + l16;                     // W_lin row == output column
  const int nrow_c = (nrow < VOCAB) ? nrow : (VOCAB - 1);
  const float* brow = W_lin + (size_t)nrow_c * EMBED + 2 * half;
  const float* arow = u + l16 * EMBED + 2 * half;     // M-tile 0; +mt*16*EMBED

  v8f acc0 = {0.f,0.f,0.f,0.f,0.f,0.f,0.f,0.f};
  v8f acc1 = acc0, acc2 = acc0, acc3 = acc0;
#pragma unroll
  for (int kk = 0; kk < EMBED / 4; ++kk) {            // 32 k-steps, 128 WMMAs
    const v2f bf = *(const v2f*)(brow + kk * 4);
    const v2f a0 = *(const v2f*)(arow + 0 * 16 * EMBED + kk * 4);
    const v2f a1 = *(const v2f*)(arow + 1 * 16 * EMBED + kk * 4);
    const v2f a2 = *(const v2f*)(arow + 2 * 16 * EMBED + kk * 4);
    const v2f a3 = *(const v2f*)(arow + 3 * 16 * EMBED + kk * 4);
    acc0 = __builtin_amdgcn_wmma_f32_16x16x4_f32(false, a0, false, bf, (short)0, acc0, false, false);
    acc1 = __builtin_amdgcn_wmma_f32_16x16x4_f32(false, a1, false, bf, (short)0, acc1, false, false);
    acc2 = __builtin_amdgcn_wmma_f32_16x16x4_f32(false, a2, false, bf, (short)0, acc2, false, false);
    acc3 = __builtin_amdgcn_wmma_f32_16x16x4_f32(false, a3, false, bf, (short)0, acc3, false, false);
  }

  if (nrow < VOCAB) {
    const float bias = b_lin[nrow];
#pragma unroll
    for (int r = 0; r < 8; ++r) {
      const int mm = r + 8 * half;
      out[(size_t)(mm +  0) * VOCAB + nrow] = acc0[r] + bias;
      out[(size_t)(mm + 16) * VOCAB + nrow] = acc1[r] + bias;
      out[(size_t)(mm + 32) * VOCAB + nrow] = acc2[r] + bias;
      out[(size_t)(mm + 48) * VOCAB + nrow] = acc3[r] + bias;
    }
  }
}

// ---------------------------------------------------------------------------
extern "C" void kernel_launch(void* const* d_in, const int* in_sizes, int n_in,
                              void* d_out, int out_size, void* d_ws, size_t ws_size,
                              hipStream_t stream) {
  (void)in_sizes; (void)n_in; (void)out_size; (void)ws_size;

  const int*   stories   = (const int*)d_in[0];
  const int*   questions = (const int*)d_in[1];
  const int*   masks     = (const int*)d_in[2];
  const float* W_A   = (const float*)d_in[3];
  const float* W_B   = (const float*)d_in[4];
  const float* W_C   = (const float*)d_in[5];
  const float* W_AT  = (const float*)d_in[6];
  const float* W_CT  = (const float*)d_in[7];
  const float* W_lin = (const float*)d_in[8];
  const float* b_lin = (const float*)d_in[9];
  float* out = (float*)d_out;

  float* ws = (float*)d_ws;
  float* m = ws;                               // 64*50*128
  float* c = m + BATCH * NSENT * EMBED;        // 64*50*128
  float* u = c + BATCH * NSENT * EMBED;        // 64*128

  k_embed_q<<<BATCH, EMBED, 0, stream>>>(questions, W_B, u);
  k_embed_mc<<<BATCH * NSENT, EMBED, 0, stream>>>(stories, masks, W_A, W_C, W_AT, W_CT, m, c);
  k_hops<<<BATCH, EMBED, 0, stream>>>(m, c, u);

  const int NT     = (VOCAB + 15) / 16;        // 3142 waves, one per N tile
  const int blocks = (NT + 7) / 8;             // 8 waves (256 thr) per block
  k_gemm<<<blocks, 256, 0, stream>>>(u, W_lin, b_lin, out);
}